// LinearAttention_26603027432181
// MI455X (gfx1250) — compile-verified
//
#include <hip/hip_runtime.h>
#include <hip/hip_bf16.h>

// Linear attention, fp32, B=4 H=16 S=8192 D=64.
// Memory-bound (~537MB @ 23.3TB/s => ~23us floor). GEMM work on
// v_wmma_f32_16x16x4_f32; staging via global_load_async_to_lds_b128
// (CDNA5 async copy, ASYNCcnt-tracked) with double buffering in pass 1.

typedef float v2f __attribute__((ext_vector_type(2)));
typedef float v8f __attribute__((ext_vector_type(8)));

static __device__ __forceinline__ v8f wmma4(v2f a, v2f b, v8f c) {
    // 8 args: (neg_a, A, neg_b, B, c_mod, C, reuse_a, reuse_b)
    return __builtin_amdgcn_wmma_f32_16x16x4_f32(
        false, a, false, b, (short)0, c, false, false);
}

// LDS byte offset of a generic pointer into __shared__ memory: flat LDS
// addresses carry the LDS offset in addr[31:0] (ISA 10.2 aperture rules).
static __device__ __forceinline__ unsigned lds_off(const void* p) {
    return (unsigned)(size_t)p;
}

// CDNA5 async copy global->LDS, 16 bytes per lane, tracked by ASYNCcnt.
static __device__ __forceinline__ void async_copy_b128(unsigned lds_byte_off,
                                                       const float* gaddr) {
    asm volatile("global_load_async_to_lds_b128 %0, %1, off"
                 :: "v"(lds_byte_off), "v"(gaddr)
                 : "memory");
}

#define WAIT_ASYNCCNT_0() asm volatile("s_wait_asynccnt 0x0" ::: "memory")
#define WAIT_ASYNCCNT_4() asm volatile("s_wait_asynccnt 0x4" ::: "memory")

#define BH    64      // B*H
#define SLEN  8192
#define DD    64
#define NCH   8       // S-chunks in pass 1
#define CHROWS (SLEN / NCH)   // 1024
#define NSTG  (CHROWS / 32)   // 32 stages of 32 rows
#define STR1  72      // pass-1 LDS row stride (2*72 % 64 == 16 -> no half conflicts)
#define STRQ  68      // pass-2 Q stride (lane banks step 4, 16 distinct)
#define STRK  72      // pass-2 kv stride

// ---------------------------------------------------------------- zero ws
__global__ __launch_bounds__(256) void zero_ws_kernel(float* __restrict__ p, int n) {
    int i = blockIdx.x * 256 + threadIdx.x;
    if (i < n) p[i] = 0.0f;
}

// ---------------------------------------------------------------- pass 1
// kv[bh][d][e] = sum_s relu(k[s,d]) * v[s,e];  ksum[bh][d] = sum_s relu(k[s,d])
__global__ __launch_bounds__(256) void la_pass1_kernel(
        const float* __restrict__ K, const float* __restrict__ V,
        float* __restrict__ ws_kv, float* __restrict__ ws_ksum) {
    __shared__ __align__(16) float lds_k[2][32 * STR1];
    __shared__ __align__(16) float lds_v[2][32 * STR1];

    const int bh    = blockIdx.x >> 3;
    const int chunk = blockIdx.x & (NCH - 1);
    const int t     = threadIdx.x;
    const int wave  = t >> 5;
    const int lane  = t & 31;
    const int half  = lane >> 4;       // K-pair select
    const int lm    = lane & 15;       // M / N index
    const int d_tile = wave & 3;                 // which 16-row d tile
    const int e_base = (wave >> 2) << 1;         // two 16-col e tiles

    const float* Kg = K + ((size_t)bh * SLEN + (size_t)chunk * CHROWS) * DD;
    const float* Vg = V + ((size_t)bh * SLEN + (size_t)chunk * CHROWS) * DD;

    // staging slots: thread t owns float4 slots t and t+256 of a 32x64 tile
    const int srow = t >> 4;            // 0..15
    const int scol = (t & 15) << 2;

    auto issue_stage = [&](int buf, int soff) {
        const float* kg = Kg + ((size_t)(soff + srow) * DD + scol);
        const float* vg = Vg + ((size_t)(soff + srow) * DD + scol);
        async_copy_b128(lds_off(&lds_k[buf][srow * STR1 + scol]), kg);
        async_copy_b128(lds_off(&lds_k[buf][(srow + 16) * STR1 + scol]), kg + 16 * DD);
        async_copy_b128(lds_off(&lds_v[buf][srow * STR1 + scol]), vg);
        async_copy_b128(lds_off(&lds_v[buf][(srow + 16) * STR1 + scol]), vg + 16 * DD);
    };

    v8f acc0 = {};
    v8f acc1 = {};
    float ksum_local = 0.0f;

    issue_stage(0, 0);                      // prologue

    for (int st = 0; st < NSTG; ++st) {
        const int cur = st & 1;
        if (st + 1 < NSTG) {
            issue_stage(1 - cur, (st + 1) * 32);   // prefetch next stage
            WAIT_ASYNCCNT_4();                     // current stage landed (in-order)
        } else {
            WAIT_ASYNCCNT_0();
        }
        __syncthreads();

        const float* lk = lds_k[cur];
        const float* lv = lds_v[cur];

        // A = relu(K)^T (M=d,K=s), B = V (K=s,N=e); 8 K-steps of 4
        #pragma unroll
        for (int ss = 0; ss < 32; ss += 4) {
            const int r0 = (ss + 2 * half) * STR1;
            const int r1 = r0 + STR1;
            v2f a;
            a.x = fmaxf(lk[r0 + d_tile * 16 + lm], 0.f);
            a.y = fmaxf(lk[r1 + d_tile * 16 + lm], 0.f);
            v2f b0, b1;
            b0.x = lv[r0 + (e_base + 0) * 16 + lm];
            b0.y = lv[r1 + (e_base + 0) * 16 + lm];
            b1.x = lv[r0 + (e_base + 1) * 16 + lm];
            b1.y = lv[r1 + (e_base + 1) * 16 + lm];
            acc0 = wmma4(a, b0, acc0);
            acc1 = wmma4(a, b1, acc1);
        }

        if (t < DD) {   // ksum partial: thread t owns column d=t
            #pragma unroll
            for (int r = 0; r < 32; ++r)
                ksum_local += fmaxf(lk[r * STR1 + t], 0.f);
        }
        __syncthreads();   // all reads of buf[cur] done before it is rewritten
    }

    // merge partials (8 chunks per bh) via f32 atomics
    float* kvp = ws_kv + (size_t)bh * (DD * DD);
    #pragma unroll
    for (int r = 0; r < 8; ++r) {
        int m = d_tile * 16 + r + 8 * half;
        atomicAdd(&kvp[m * DD + (e_base + 0) * 16 + lm], acc0[r]);
        atomicAdd(&kvp[m * DD + (e_base + 1) * 16 + lm], acc1[r]);
    }
    if (t < DD) atomicAdd(&ws_ksum[bh * DD + t], ksum_local);
}

// ---------------------------------------------------------------- pass 2
// out[s,e] = (phi(q)[s,:] @ kv)[e] / max(phi(q)[s,:] . ksum, 1e-6)
__global__ __launch_bounds__(256) void la_pass2_kernel(
        const float* __restrict__ Q, const float* __restrict__ ws_kv,
        const float* __restrict__ ws_ksum, float* __restrict__ out) {
    __shared__ __align__(16) float lds_q[128 * STRQ];
    __shared__ __align__(16) float lds_kv[DD * STRK];
    __shared__ float lds_ks[DD];
    __shared__ float lds_norm[128];

    const int bh   = blockIdx.x >> 6;     // 8192/128 = 64 s-blocks per bh
    const int sblk = blockIdx.x & 63;
    const int t    = threadIdx.x;
    const int wave = t >> 5;
    const int lane = t & 31;
    const int half = lane >> 4;
    const int lm   = lane & 15;

    const float* Qg  = Q + ((size_t)bh * SLEN + (size_t)sblk * 128) * DD;
    const float* kvg = ws_kv + (size_t)bh * (DD * DD);

    // stage kv (64x64) and raw Q (128x64) via async copies
    #pragma unroll
    for (int i = 0; i < 4; ++i) {
        int j = t + i * 256;              // 1024 float4 slots
        int r_ = j >> 4, c_ = (j & 15) << 2;
        async_copy_b128(lds_off(&lds_kv[r_ * STRK + c_]), kvg + r_ * DD + c_);
    }
    #pragma unroll
    for (int i = 0; i < 8; ++i) {
        int j = t + i * 256;              // 2048 float4 slots
        int r_ = j >> 4, c_ = (j & 15) << 2;
        async_copy_b128(lds_off(&lds_q[r_ * STRQ + c_]), Qg + (size_t)r_ * DD + c_);
    }
    if (t < DD) lds_ks[t] = ws_ksum[bh * DD + t];
    WAIT_ASYNCCNT_0();
    __syncthreads();

    // wave w owns rows [w*16, w*16+16) x all 64 cols: 4 accumulators
    v8f acc[4] = {};
    #pragma unroll
    for (int d0 = 0; d0 < DD; d0 += 4) {
        v2f a;
        const int qrow = (wave * 16 + lm) * STRQ + d0 + 2 * half;
        a.x = fmaxf(lds_q[qrow + 0], 0.f);      // relu at fragment build
        a.y = fmaxf(lds_q[qrow + 1], 0.f);
        const int r0 = (d0 + 2 * half) * STRK;
        #pragma unroll
        for (int et = 0; et < 4; ++et) {
            v2f b;
            b.x = lds_kv[r0 + et * 16 + lm];
            b.y = lds_kv[r0 + STRK + et * 16 + lm];
            acc[et] = wmma4(a, b, acc[et]);
        }
    }

    // normalizer: lanes 0..15 each own one row of the wave's tile
    if (half == 0) {
        const int row = wave * 16 + lm;
        float s = 0.0f;
        #pragma unroll 8
        for (int d = 0; d < DD; ++d)
            s += fmaxf(lds_q[row * STRQ + d], 0.f) * lds_ks[d];
        lds_norm[row] = fmaxf(s, 1e-6f);
    }
    __syncthreads();

    float* og = out + ((size_t)bh * SLEN + (size_t)sblk * 128) * DD;
    #pragma unroll
    for (int r = 0; r < 8; ++r) {
        const int m   = wave * 16 + r + 8 * half;   // row in s-block
        const float inv = 1.0f / lds_norm[m];
        #pragma unroll
        for (int et = 0; et < 4; ++et) {
            og[(size_t)m * DD + et * 16 + lm] = acc[et][r] * inv;
        }
    }
}

// ---------------------------------------------------------------- launch
extern "C" void kernel_launch(void* const* d_in, const int* in_sizes, int n_in,
                              void* d_out, int out_size, void* d_ws, size_t ws_size,
                              hipStream_t stream) {
    (void)in_sizes; (void)n_in; (void)out_size; (void)ws_size;
    const float* q = (const float*)d_in[0];
    const float* k = (const float*)d_in[1];
    const float* v = (const float*)d_in[2];
    float* out     = (float*)d_out;

    float* ws_kv   = (float*)d_ws;                 // [BH][64][64]
    float* ws_ksum = ws_kv + (size_t)BH * DD * DD; // [BH][64]

    const int nz = BH * DD * DD + BH * DD;         // 266,240 floats (~1.0 MB)
    zero_ws_kernel<<<(nz + 255) / 256, 256, 0, stream>>>(ws_kv, nz);
    la_pass1_kernel<<<BH * NCH, 256, 0, stream>>>(k, v, ws_kv, ws_ksum);
    la_pass2_kernel<<<BH * (SLEN / 128), 256, 0, stream>>>(q, ws_kv, ws_ksum, out);
}